// LatticeKTAP_49667001811213
// MI455X (gfx1250) — compile-verified
//
#include <hip/hip_runtime.h>
#include <hip/hip_bf16.h>

typedef __attribute__((ext_vector_type(2))) float v2f;
typedef __attribute__((ext_vector_type(8))) float v8f;

#define TEACHER_DIM 512
#define STUDENT_DIM 128
#define N_TASKS     8
#define BATCH       16384

#define ROWS_PER_BLOCK 128   // 8 waves * 16 rows
#define KC             64    // K-chunk staged in LDS (double buffered)
#define PAIR_STRIDE    288   // dwords per K-pair row: 256 data + 32 pad
                             // (288 % 64 == 32 -> half-waves hit disjoint bank halves)
#define STAGE_OPS      32    // async b32 copies per thread per chunk (64*128 / 256)

#if __has_builtin(__builtin_amdgcn_global_load_async_to_lds_b32)
#define HAVE_ASYNC_LDS 1
#else
#define HAVE_ASYNC_LDS 0
#endif

#if HAVE_ASYNC_LDS
typedef __attribute__((address_space(1))) int g_i32;
typedef __attribute__((address_space(3))) int l_i32;
#endif

template <int N>
__device__ __forceinline__ void wait_async_le() {
#if __has_builtin(__builtin_amdgcn_s_wait_asynccnt)
    __builtin_amdgcn_s_wait_asynccnt(N);
#else
    asm volatile("s_wait_asynccnt %0" ::"i"(N) : "memory");
#endif
}

// Stage W[kc:kc+KC, 0:128] into LDS, K-pair interleaved:
//   element (k, n) -> lbuf[(k>>1)*PAIR_STRIDE + 2*n + (k&1)]
// so each WMMA B fragment {W[k][n], W[k+1][n]} is one aligned b64.
__device__ __forceinline__ void stage_chunk_async(const float* __restrict__ Wc,
                                                  float* __restrict__ lbuf, int tid) {
#if HAVE_ASYNC_LDS
#pragma unroll
    for (int i = 0; i < STAGE_OPS; ++i) {
        const int idx = tid + i * 256;        // 0..8191
        const int k   = idx >> 7;             // 0..63
        const int n   = idx & 127;            // 0..127
        __builtin_amdgcn_global_load_async_to_lds_b32(
            (g_i32*)(Wc + (size_t)k * STUDENT_DIM + n),
            (l_i32*)(lbuf + (k >> 1) * PAIR_STRIDE + 2 * n + (k & 1)),
            /*offset=*/0, /*cpol=*/0);
    }
#endif
}

__device__ __forceinline__ void stage_chunk_sync(const float* __restrict__ Wc,
                                                 float* __restrict__ lbuf, int tid) {
#pragma unroll
    for (int i = 0; i < STAGE_OPS; ++i) {
        const int idx = tid + i * 256;
        const int k   = idx >> 7;
        const int n   = idx & 127;
        lbuf[(k >> 1) * PAIR_STRIDE + 2 * n + (k & 1)] = Wc[(size_t)k * STUDENT_DIM + n];
    }
}

// ---------------------------------------------------------------------------
// teacher_feats = (hit ? cache_emb[slot] : 0) @ W   via V_WMMA_F32_16X16X4_F32
// ---------------------------------------------------------------------------
__global__ __launch_bounds__(256) void lattice_feats_kernel(
    const float* __restrict__ cache_emb,     // (CACHE, 512)
    const float* __restrict__ W,             // (512, 128)
    const int*   __restrict__ slot_idx,      // (BATCH)
    const unsigned char* __restrict__ hit,   // (BATCH) bool (1 byte)
    float* __restrict__ out_feats)           // (BATCH, 128)
{
    __shared__ float ldsW[2][(KC / 2) * PAIR_STRIDE];

    const int tid  = threadIdx.x;
    const int lane = tid & 31;
    const int wave = tid >> 5;
    const int nl   = lane & 15;          // N (or M) index within tile
    const int h    = lane >> 4;          // half-wave
    const int ksel = h << 1;             // half-wave selects K pair {0,1} or {2,3}

    // Row this lane gathers for the A fragment (lane%16 == M index)
    const int row = blockIdx.x * ROWS_PER_BLOCK + wave * 16 + nl;
    const long long slot = (long long)slot_idx[row];
    const float maskf = hit[row] ? 1.0f : 0.0f;
    const float* __restrict__ arow = cache_emb + slot * TEACHER_DIM;

    v8f acc[8];
#pragma unroll
    for (int t = 0; t < 8; ++t) acc[t] = (v8f){0.f,0.f,0.f,0.f,0.f,0.f,0.f,0.f};

    // Depth-2 A-fragment pipeline: prefetch k and k+4 before the loop,
    // inside the loop prefetch k+8 so the global load latency overlaps a
    // full k-step of DS+WMMA work.
    float2 a0 = *(const float2*)(arow + 0 + ksel);
    float2 a1 = *(const float2*)(arow + 4 + ksel);
    int gk = 0;

#if HAVE_ASYNC_LDS
    stage_chunk_async(W, ldsW[0], tid);   // prefetch chunk 0
#endif

    const int NCHUNK = TEACHER_DIM / KC;  // 8
    for (int c = 0; c < NCHUNK; ++c) {
        float* __restrict__ lbuf = ldsW[c & 1];

#if HAVE_ASYNC_LDS
        // Issue next chunk into the other buffer (its readers passed the
        // trailing barrier of iteration c-1), wait for this chunk's copies
        // (async loads complete in order), then barrier for visibility.
        if (c + 1 < NCHUNK) {
            stage_chunk_async(W + (size_t)(c + 1) * KC * STUDENT_DIM, ldsW[(c + 1) & 1], tid);
            wait_async_le<STAGE_OPS>();
        } else {
            wait_async_le<0>();
        }
        __syncthreads();
#else
        __syncthreads();
        stage_chunk_sync(W + (size_t)c * KC * STUDENT_DIM, lbuf, tid);
        __syncthreads();
#endif

#pragma unroll 4
        for (int kk = 0; kk < KC; kk += 4) {
            // A fragment: 16x4 f32. Masked rows contribute zero (v_pk_mul).
            v2f av;
            av.x = a0.x * maskf;
            av.y = a0.y * maskf;
            a0 = a1;
            int knext = gk + 8;
            if (knext > TEACHER_DIM - 4) knext = TEACHER_DIM - 4;  // safe dummy reload
            a1 = *(const float2*)(arow + knext + ksel);
            gk += 4;

            // B fragments: one aligned ds_load_b64 per tile, zero shuffles.
            const float* __restrict__ bbase =
                &lbuf[((kk >> 1) + h) * PAIR_STRIDE + 2 * nl];
#pragma unroll
            for (int t = 0; t < 8; ++t) {
                const v2f bv = *(const v2f*)(bbase + 32 * t);
                acc[t] = __builtin_amdgcn_wmma_f32_16x16x4_f32(
                    /*neg_a=*/false, av, /*neg_b=*/false, bv,
                    /*c_mod=*/(short)0, acc[t],
                    /*reuse_a=*/false, /*reuse_b=*/false);
            }
        }
        __syncthreads();  // all waves done reading lbuf before it is restaged
    }

    // C/D layout: VGPR j -> M=j (lanes 0-15) or M=j+8 (lanes 16-31), N = lane%16
    const int mbase = h << 3;
    float* __restrict__ obase =
        out_feats + (size_t)(blockIdx.x * ROWS_PER_BLOCK + wave * 16 + mbase) * STUDENT_DIM + nl;
#pragma unroll
    for (int j = 0; j < 8; ++j) {
#pragma unroll
        for (int t = 0; t < 8; ++t) {
            obase[(size_t)j * STUDENT_DIM + t * 16] = acc[t][j];
        }
    }
}

// ---------------------------------------------------------------------------
// teacher_logits = hit ? cache_logits[slot] : 0   (one row of 8 f32 per thread)
// ---------------------------------------------------------------------------
__global__ __launch_bounds__(256) void lattice_logits_kernel(
    const float* __restrict__ cache_logits,  // (CACHE, 8)
    const int*   __restrict__ slot_idx,
    const unsigned char* __restrict__ hit,
    float* __restrict__ out_logits)          // (BATCH, 8)
{
    const int r = blockIdx.x * blockDim.x + threadIdx.x;   // 0..BATCH-1
    float4 a = {0.f, 0.f, 0.f, 0.f};
    float4 b = {0.f, 0.f, 0.f, 0.f};
    if (hit[r]) {
        const float4* src = (const float4*)(cache_logits + (size_t)slot_idx[r] * N_TASKS);
        a = src[0];
        b = src[1];
    }
    float4* dst = (float4*)(out_logits + (size_t)r * N_TASKS);
    dst[0] = a;
    dst[1] = b;
}

// ---------------------------------------------------------------------------
extern "C" void kernel_launch(void* const* d_in, const int* in_sizes, int n_in,
                              void* d_out, int out_size, void* d_ws, size_t ws_size,
                              hipStream_t stream) {
    const float* cache_emb    = (const float*)d_in[0];
    const float* cache_logits = (const float*)d_in[1];
    const float* W            = (const float*)d_in[2];
    const int*   slot_idx     = (const int*)d_in[3];
    const unsigned char* hit  = (const unsigned char*)d_in[4];  // jnp bool -> 1 byte

    float* out_feats  = (float*)d_out;
    float* out_logits = (float*)d_out + (size_t)BATCH * STUDENT_DIM;

    lattice_feats_kernel<<<BATCH / ROWS_PER_BLOCK, 256, 0, stream>>>(
        cache_emb, W, slot_idx, hit, out_feats);
    lattice_logits_kernel<<<BATCH / 256, 256, 0, stream>>>(
        cache_logits, slot_idx, hit, out_logits);
}